// GMMOutlierDetector_24927990186092
// MI455X (gfx1250) — compile-verified
//
#include <hip/hip_runtime.h>
#include <math.h>

// Problem constants (from reference setup_inputs)
#define BATCH 8
#define CCH   768
#define HH    32
#define WW    64
#define HW    (HH*WW)          // 2048
#define NPTS  (BATCH*HH*WW)    // 16384
#define DD    768
#define KK    19
#define MMIX  5
#define KM    95
#define KMP   96                // padded to 6 tiles of 16
#define LOG2PI 1.8378770664093453f

typedef __attribute__((ext_vector_type(2))) float v2f;
typedef __attribute__((ext_vector_type(8))) float v8f;

// Explicit global-address-space pointers so the backend emits global_load_*
// (not flat_load_*, which ties up DScnt + the LDS path per ISA §10.1).
typedef const float __attribute__((address_space(1)))* gcf;
typedef const v2f   __attribute__((address_space(1)))* gcv2;

// ---------------------------------------------------------------------------
// Prep kernel 1: build pair-interleaved B matrices (D x 96, padded cols = 0)
//   b1[(d>>1)*192 + c*2 + (d&1)] = exp(-log_var[c][d])            (iv)
//   b2[  same index            ] = -2 * mean[c][d] * iv
// Pair interleave lets the GEMM load each B fragment as one b64 per lane.
// ---------------------------------------------------------------------------
__global__ __launch_bounds__(256) void gmm_prep_b(
    const float* __restrict__ means, const float* __restrict__ log_vars,
    float* __restrict__ b1, float* __restrict__ b2) {
  int idx = blockIdx.x * 256 + threadIdx.x;      // over 96*768 elements
  if (idx >= KMP * DD) return;
  int d = idx / KMP;
  int c = idx % KMP;
  float v1 = 0.0f, v2 = 0.0f;
  if (c < KM) {
    float lv = log_vars[c * DD + d];
    float mu = means[c * DD + d];
    float iv = expf(-lv);
    v1 = iv;
    v2 = -2.0f * mu * iv;
  }
  int o = (d >> 1) * (2 * KMP) + c * 2 + (d & 1);
  b1[o] = v1;
  b2[o] = v2;
}

// ---------------------------------------------------------------------------
// Prep kernel 2: per-column constant
//   c_km = log_w - 0.5*(D*log2pi + sum(log_var) + sum(mu^2*iv))
// Padded columns get -1e30 so exp() kills them in the logsumexp.
// ---------------------------------------------------------------------------
__global__ __launch_bounds__(128) void gmm_prep_c(
    const float* __restrict__ means, const float* __restrict__ log_vars,
    const float* __restrict__ log_w, float* __restrict__ ckm) {
  int c = threadIdx.x;
  if (c >= KMP) return;
  if (c >= KM) { ckm[c] = -1.0e30f; return; }
  float slv = 0.0f, smi = 0.0f;
  for (int d = 0; d < DD; ++d) {
    float lv = log_vars[c * DD + d];
    float mu = means[c * DD + d];
    slv += lv;
    smi += mu * mu * expf(-lv);
  }
  ckm[c] = log_w[c] - 0.5f * ((float)DD * LOG2PI + slv + smi);
}

// ---------------------------------------------------------------------------
// Main kernel: fused dual-GEMM (WMMA f32 16x16x4) + logsumexp/softmax epilogue
// Block = 128 threads = 4 waves; each wave: 2 row-tiles (32 pts) x 6 col-tiles.
// ---------------------------------------------------------------------------
__global__ __launch_bounds__(128) void gmm_main(
    const float* __restrict__ feats,
    const float* __restrict__ b1, const float* __restrict__ b2,
    const float* __restrict__ ckm, float* __restrict__ out) {
  __shared__ float sdata[128 * KMP];   // 48 KB: per-point maha rows
  __shared__ float cksh[KMP];

  const int tid  = threadIdx.x;
  const int wave = tid >> 5;
  const int lane = tid & 31;
  const int lrow = lane & 15;   // M (A) / N (B,C) index within half-wave
  const int lhi  = lane >> 4;   // selects K pair (A/B) or M+8 (C/D)

  if (tid < KMP) cksh[tid] = ckm[tid];

  // 12 independent accumulator chains
  v8f acc[2][6];
#pragma unroll
  for (int nt = 0; nt < 2; ++nt)
#pragma unroll
    for (int ct = 0; ct < 6; ++ct)
      acc[nt][ct] = (v8f){0.f,0.f,0.f,0.f,0.f,0.f,0.f,0.f};

  // A-fragment lane pointers. x[n][d] = feats[(n>>11)*C*HW + d*HW + (n&2047)]
  // A 16x4 layout: lanes 0-15 hold K={0,1}, lanes 16-31 hold K={2,3}, M=lane&15
  gcf ap[2];
#pragma unroll
  for (int nt = 0; nt < 2; ++nt) {
    int n0  = blockIdx.x * 128 + wave * 32 + nt * 16;
    int bb  = n0 >> 11;
    int rem = n0 & 2047;
    ap[nt] = (gcf)(feats + (size_t)bb * (size_t)(CCH * HW)
                         + (size_t)(2 * lhi) * HW + rem + lrow);
  }

  // B-fragment lane pointers (pair-interleaved layout, one b64 per fragment)
  gcf bp1[6];
  gcf bp2[6];
#pragma unroll
  for (int ct = 0; ct < 6; ++ct) {
    int off = lhi * (2 * KMP) + (ct * 16 + lrow) * 2;
    bp1[ct] = (gcf)(b1 + off);
    bp2[ct] = (gcf)(b2 + off);
  }

  // Main D loop: 192 iterations, 24 WMMAs each
  for (int d0 = 0; d0 < DD; d0 += 4) {
    v2f a[2], asq[2];
#pragma unroll
    for (int nt = 0; nt < 2; ++nt) {
      float a0 = ap[nt][0];
      float a1 = ap[nt][HW];
      ap[nt] += 4 * HW;
      a[nt].x = a0;        a[nt].y = a1;
      asq[nt].x = a0 * a0; asq[nt].y = a1 * a1;
    }
#pragma unroll
    for (int ct = 0; ct < 6; ++ct) {
      v2f vb1 = *(gcv2)bp1[ct];
      v2f vb2 = *(gcv2)bp2[ct];
      bp1[ct] += 2 * (2 * KMP);
      bp2[ct] += 2 * (2 * KMP);
#pragma unroll
      for (int nt = 0; nt < 2; ++nt) {
        acc[nt][ct] = __builtin_amdgcn_wmma_f32_16x16x4_f32(
            false, asq[nt], false, vb1, (short)0, acc[nt][ct], false, false);
        acc[nt][ct] = __builtin_amdgcn_wmma_f32_16x16x4_f32(
            false, a[nt],   false, vb2, (short)0, acc[nt][ct], false, false);
      }
    }
  }

  // Spill accumulators to LDS per documented C/D layout:
  // VGPR r: lanes 0-15 -> M=r, lanes 16-31 -> M=r+8; N = lane&15
#pragma unroll
  for (int nt = 0; nt < 2; ++nt)
#pragma unroll
    for (int ct = 0; ct < 6; ++ct)
#pragma unroll
      for (int r = 0; r < 8; ++r) {
        int pl = wave * 32 + nt * 16 + r + 8 * lhi;
        sdata[pl * KMP + ct * 16 + lrow] = acc[nt][ct][r];
      }
  __syncthreads();

  // Fused epilogue: one thread per point
  {
    const float* row = &sdata[tid * KMP];
    float lse[KK];
#pragma unroll
    for (int k = 0; k < KK; ++k) {
      float lp[MMIX];
      float mx = -3.0e38f;
#pragma unroll
      for (int m = 0; m < MMIX; ++m) {
        int c = k * MMIX + m;
        lp[m] = -0.5f * row[c] + cksh[c];
        mx = fmaxf(mx, lp[m]);
      }
      float s = 0.0f;
#pragma unroll
      for (int m = 0; m < MMIX; ++m) s += expf(lp[m] - mx);
      lse[k] = mx + logf(s);
    }
    float mv = lse[0];
#pragma unroll
    for (int k = 1; k < KK; ++k) mv = fmaxf(mv, lse[k]);
    float s = 0.0f;
#pragma unroll
    for (int k = 0; k < KK; ++k) s += expf(lse[k] - mv);
    // max softmax prob = 1/s  ->  anomaly = 1 - 1/s
    out[blockIdx.x * 128 + tid] = 1.0f - 1.0f / s;
  }
}

// ---------------------------------------------------------------------------
extern "C" void kernel_launch(void* const* d_in, const int* in_sizes, int n_in,
                              void* d_out, int out_size, void* d_ws, size_t ws_size,
                              hipStream_t stream) {
  const float* feats     = (const float*)d_in[0];  // (8,768,32,64)
  const float* means     = (const float*)d_in[1];  // (19,5,768)
  const float* log_vars  = (const float*)d_in[2];  // (19,5,768)
  const float* log_w     = (const float*)d_in[3];  // (19,5)
  float* out = (float*)d_out;                      // (8,32,64) = 16384

  // Workspace layout (floats): b1 [768*96], b2 [768*96], ckm [96]
  float* b1  = (float*)d_ws;
  float* b2  = b1 + KMP * DD;
  float* ckm = b2 + KMP * DD;

  (void)in_sizes; (void)n_in; (void)out_size; (void)ws_size;

  int prep_elems = KMP * DD;
  gmm_prep_b<<<(prep_elems + 255) / 256, 256, 0, stream>>>(means, log_vars, b1, b2);
  gmm_prep_c<<<1, 128, 0, stream>>>(means, log_vars, log_w, ckm);
  gmm_main<<<NPTS / 128, 128, 0, stream>>>(feats, b1, b2, ckm, out);
}